// ModelMTGCAIN_12893491823112
// MI455X (gfx1250) — compile-verified
//
#include <hip/hip_runtime.h>
#include <hip/hip_bf16.h>

typedef __attribute__((ext_vector_type(2))) float v2f;
typedef __attribute__((ext_vector_type(8))) float v8f;

#define DIM 128
#define BN_EPS 1e-5f

__device__ __forceinline__ void async_g2l_b128(uint32_t lds_byte_addr, uint64_t gaddr) {
    // GLOBAL_LOAD_ASYNC_TO_LDS_B128 (GV mode): per-lane 16B global -> LDS,
    // tracked with ASYNCcnt. Bypasses VGPRs entirely.
    asm volatile("global_load_async_to_lds_b128 %0, %1, off"
                 :: "v"(lds_byte_addr), "v"(gaddr) : "memory");
}

__device__ __forceinline__ void wait_asynccnt0() {
#if __has_builtin(__builtin_amdgcn_s_wait_asynccnt)
    __builtin_amdgcn_s_wait_asynccnt(0);
#else
    asm volatile("s_wait_asynccnt 0" ::: "memory");
#endif
}

// ---------------------------------------------------------------------------
// K0: agg = epsilon * v  (epsilon read on-device), zero stats block (1024 f)
// ---------------------------------------------------------------------------
__global__ __launch_bounds__(256)
void init_kernel(const float* __restrict__ v, const float* __restrict__ eps,
                 float* __restrict__ agg, float* __restrict__ stats, int total4) {
    int i = blockIdx.x * blockDim.x + threadIdx.x;
    if (i < 256) {               // 1024 stat floats = 256 float4
        float4 z = make_float4(0.f, 0.f, 0.f, 0.f);
        reinterpret_cast<float4*>(stats)[i] = z;
    }
    if (i < total4) {
        float e = eps[0];
        float4 x = reinterpret_cast<const float4*>(v)[i];
        float4 y = make_float4(e * x.x, e * x.y, e * x.z, e * x.w);
        reinterpret_cast<float4*>(agg)[i] = y;
    }
}

// ---------------------------------------------------------------------------
// K1: SpMM scatter. One wave32 per edge; each lane moves a float4 of the
// 128-wide feature row (gather from L2-resident v, 4 f32 atomics to agg).
// ---------------------------------------------------------------------------
__global__ __launch_bounds__(256)
void spmm_kernel(const float* __restrict__ v, const float* __restrict__ ev,
                 const int* __restrict__ er, const int* __restrict__ ec,
                 float* __restrict__ agg, int E) {
    int e = blockIdx.x * (blockDim.x >> 5) + (threadIdx.x >> 5);
    if (e >= E) return;
    int lane = threadIdx.x & 31;
    int row = er[e];
    int col = ec[e];
    float w = ev[e];
    const float4 x = *reinterpret_cast<const float4*>(v + col * DIM + lane * 4);
    float* dst = agg + row * DIM + lane * 4;
    atomicAdd(dst + 0, w * x.x);
    atomicAdd(dst + 1, w * x.y);
    atomicAdd(dst + 2, w * x.z);
    atomicAdd(dst + 3, w * x.w);
}

// ---------------------------------------------------------------------------
// K2/K4: P = op(X) @ W + bias, fused column sum / sumsq for BatchNorm stats.
// op(X) = identity (layer 1) or relu(X*scaleA + shiftA) (BN1+ReLU fused into
// the A-operand staging for layer 2).
// Block = 128 threads = 4 wave32s; block covers a 64-row slab; each wave
// computes a 16x128 strip as 8 accumulators via v_wmma_f32_16x16x4_f32.
//
// W is staged into LDS K-pair-interleaved: LWP[(k/2)*256 + 2n + (k&1)], so
// each WMMA B fragment (W[k][n], W[k+1][n]) is one aligned ds_load_b64 that
// lands directly in a WMMA-ready even register pair (no repack movs).
// The X slab goes global->LDS via async b128 copies (ASYNCcnt) when no
// per-element transform is needed and the slab is fully in range.
// ---------------------------------------------------------------------------
template <bool BN_ON_A>
__global__ __launch_bounds__(128)
void gemm128_kernel(const float* __restrict__ X,    // N x 128
                    const float* __restrict__ W,    // 128 x 128 row-major
                    const float* __restrict__ bias, // 128
                    const float* __restrict__ scA,  // 128 (BN scale on A) or null
                    const float* __restrict__ shA,  // 128 (BN shift on A) or null
                    float* __restrict__ P,          // N x 128 (pre-BN output)
                    float* __restrict__ colsum,     // 128 global accumulators
                    float* __restrict__ colsumsq,   // 128
                    int N) {
    __shared__ float LWP[DIM * DIM]; // 64 KB weights, K-pair interleaved
    __shared__ float LX[64][DIM];    // 32 KB A slab
    __shared__ float cs[DIM];
    __shared__ float cq[DIM];

    const int tid  = threadIdx.x;   // 0..127
    const int lane = tid & 31;
    const int wv   = tid >> 5;      // wave 0..3
    const int half = lane >> 4;     // 0,1
    const int mn   = lane & 15;     // 0..15
    const int rowBase = blockIdx.x * 64;

    cs[tid] = 0.f;
    cq[tid] = 0.f;

    // ---- Stage W into LDS, interleaving K pairs:
    //      LWP[kp*256 + 2n + j] = W[2kp+j][n]. Task = (kp 0..63, n4 0..31).
    for (int idx = tid; idx < 64 * 32; idx += 128) {
        const int kp = idx >> 5;
        const int n4 = idx & 31;
        const float4 r0 = *reinterpret_cast<const float4*>(W + (2 * kp + 0) * DIM + n4 * 4);
        const float4 r1 = *reinterpret_cast<const float4*>(W + (2 * kp + 1) * DIM + n4 * 4);
        float* dst = &LWP[kp * 256 + n4 * 8];
        *reinterpret_cast<float4*>(dst + 0) = make_float4(r0.x, r1.x, r0.y, r1.y);
        *reinterpret_cast<float4*>(dst + 4) = make_float4(r0.z, r1.z, r0.w, r1.w);
    }

    // ---- Stage 64x128 X slab. Async path when no transform and fully in
    //      range (block-uniform condition); VGPR path otherwise. ----
    const bool fullSlab = (rowBase + 64 <= N);
    if (!BN_ON_A && fullSlab) {
        const uint32_t lxbase = (uint32_t)(uintptr_t)(&LX[0][0]);
        const float* src = X + (size_t)rowBase * DIM;
        for (int i = tid; i < 64 * DIM / 4; i += 128) {
            async_g2l_b128(lxbase + (uint32_t)i * 16u,
                           (uint64_t)(uintptr_t)(src + i * 4));
        }
    } else {
        for (int i = tid; i < 64 * 32; i += 128) {   // i indexes float4
            int r  = i >> 5;
            int c4 = (i & 31) << 2;
            int gr = rowBase + r;
            float4 x = make_float4(0.f, 0.f, 0.f, 0.f);
            if (gr < N) {
                x = *reinterpret_cast<const float4*>(X + gr * DIM + c4);
                if (BN_ON_A) {
                    x.x = fmaxf(fmaf(x.x, scA[c4 + 0], shA[c4 + 0]), 0.f);
                    x.y = fmaxf(fmaf(x.y, scA[c4 + 1], shA[c4 + 1]), 0.f);
                    x.z = fmaxf(fmaf(x.z, scA[c4 + 2], shA[c4 + 2]), 0.f);
                    x.w = fmaxf(fmaf(x.w, scA[c4 + 3], shA[c4 + 3]), 0.f);
                }
            }
            *reinterpret_cast<float4*>(&LX[r][c4]) = x;
        }
    }
    wait_asynccnt0();
    __syncthreads();

    v8f acc[8] = {};
    const int arow = wv * 16 + mn;       // A-fragment slab row (lane = M)

    #pragma unroll 4
    for (int k0 = 0; k0 < DIM; k0 += 4) {
        // A fragment: lane = M, two consecutive K values for this lane-half.
        v2f a;
        a.x = LX[arow][k0 + half * 2 + 0];
        a.y = LX[arow][k0 + half * 2 + 1];
        const int kp = (k0 >> 1) + half;     // K-pair index for B
        #pragma unroll
        for (int nt = 0; nt < 8; ++nt) {
            const int n = nt * 16 + mn;
            const v2f b = *reinterpret_cast<const v2f*>(&LWP[kp * 256 + n * 2]);
            acc[nt] = __builtin_amdgcn_wmma_f32_16x16x4_f32(
                false, a, false, b, (short)0, acc[nt], false, false);
        }
    }

    // Epilogue: bias add, store pre-BN result, accumulate column stats.
    #pragma unroll
    for (int nt = 0; nt < 8; ++nt) {
        const int n = nt * 16 + mn;
        const float bn = bias[n];
        float s = 0.f, q = 0.f;
        #pragma unroll
        for (int r = 0; r < 8; ++r) {
            const int m  = wv * 16 + half * 8 + r;   // C/D layout: M = r + 8*half
            const int gr = rowBase + m;
            float val = acc[nt][r] + bn;
            if (gr < N) {
                P[gr * DIM + n] = val;
                s += val;
                q += val * val;
            }
        }
        atomicAdd(&cs[n], s);
        atomicAdd(&cq[n], q);
    }
    __syncthreads();
    atomicAdd(&colsum[tid],   cs[tid]);
    atomicAdd(&colsumsq[tid], cq[tid]);
}

// ---------------------------------------------------------------------------
// K3/K5: fold BN stats into per-column scale/shift.
// ---------------------------------------------------------------------------
__global__ __launch_bounds__(128)
void bn_finalize_kernel(const float* __restrict__ sum, const float* __restrict__ sumsq,
                        const float* __restrict__ gamma, const float* __restrict__ beta,
                        float* __restrict__ scale, float* __restrict__ shift, float invN) {
    int j = threadIdx.x;
    float mean = sum[j] * invN;
    float var  = sumsq[j] * invN - mean * mean;
    float sc   = gamma[j] * rsqrtf(var + BN_EPS);
    scale[j] = sc;
    shift[j] = beta[j] - mean * sc;
}

// ---------------------------------------------------------------------------
// K6: out = relu(out * scale2[col] + shift2[col]) in place.
// ---------------------------------------------------------------------------
__global__ __launch_bounds__(256)
void bn_relu_inplace_kernel(float* __restrict__ out, const float* __restrict__ scale,
                            const float* __restrict__ shift, int total) {
    int i = blockIdx.x * blockDim.x + threadIdx.x;
    if (i < total) {
        int c = i & (DIM - 1);
        out[i] = fmaxf(fmaf(out[i], scale[c], shift[c]), 0.f);
    }
}

// ---------------------------------------------------------------------------
extern "C" void kernel_launch(void* const* d_in, const int* in_sizes, int n_in,
                              void* d_out, int out_size, void* d_ws, size_t ws_size,
                              hipStream_t stream) {
    const float* v        = (const float*)d_in[0];
    const float* edge_val = (const float*)d_in[1];
    const float* epsilon  = (const float*)d_in[2];
    const float* w1       = (const float*)d_in[3];
    const float* b1       = (const float*)d_in[4];
    const float* gamma1   = (const float*)d_in[5];
    const float* beta1    = (const float*)d_in[6];
    const float* w2       = (const float*)d_in[7];
    const float* b2       = (const float*)d_in[8];
    const float* gamma2   = (const float*)d_in[9];
    const float* beta2    = (const float*)d_in[10];
    const int*   edge_row = (const int*)d_in[11];
    const int*   edge_col = (const int*)d_in[12];
    float* out = (float*)d_out;

    const int N = in_sizes[0] / DIM;
    const int E = in_sizes[1];

    float* ws    = (float*)d_ws;
    float* agg   = ws;                     // N*128
    float* hpre  = ws + (size_t)N * DIM;   // N*128
    float* stats = ws + (size_t)2 * N * DIM;
    float* h_sum   = stats +   0;
    float* h_sumsq = stats + 128;
    float* o_sum   = stats + 256;
    float* o_sumsq = stats + 384;
    float* scale1  = stats + 512;
    float* shift1  = stats + 640;
    float* scale2  = stats + 768;
    float* shift2  = stats + 896;

    const int total  = N * DIM;
    const int total4 = total / 4;
    const float invN = 1.0f / (float)N;

    // K0: agg = eps * v ; zero stats
    init_kernel<<<(total4 + 255) / 256, 256, 0, stream>>>(v, epsilon, agg, stats, total4);

    // K1: agg += A @ v  (COO scatter, wave per edge)
    spmm_kernel<<<(E + 7) / 8, 256, 0, stream>>>(v, edge_val, edge_row, edge_col, agg, E);

    // K2: hpre = agg @ w1 + b1, accumulate BN1 stats
    const int gBlocks = (N + 63) / 64;
    gemm128_kernel<false><<<gBlocks, 128, 0, stream>>>(
        agg, w1, b1, nullptr, nullptr, hpre, h_sum, h_sumsq, N);

    // K3: BN1 scale/shift
    bn_finalize_kernel<<<1, 128, 0, stream>>>(h_sum, h_sumsq, gamma1, beta1, scale1, shift1, invN);

    // K4: out_pre = relu(bn1(hpre)) @ w2 + b2 (BN1+ReLU fused into A staging),
    //     accumulate BN2 stats; write to d_out.
    gemm128_kernel<true><<<gBlocks, 128, 0, stream>>>(
        hpre, w2, b2, scale1, shift1, out, o_sum, o_sumsq, N);

    // K5: BN2 scale/shift
    bn_finalize_kernel<<<1, 128, 0, stream>>>(o_sum, o_sumsq, gamma2, beta2, scale2, shift2, invN);

    // K6: out = relu(bn2(out)) in place
    bn_relu_inplace_kernel<<<(total + 255) / 256, 256, 0, stream>>>(out, scale2, shift2, total);
}